// CTCLoss_24386824306855
// MI455X (gfx1250) — compile-verified
//
#include <hip/hip_runtime.h>
#include <stdint.h>

// Problem constants (match the reference).
#define B_   32
#define T_   2048
#define V_   1024
#define S_   256
#define L_   513            // 2*S + 1 extended states
#define NTH  512            // 16 wave32's; thread 0 owns states 0 AND 512 (both blank)
#define NBUF 4              // async prefetch ring depth (power of 2)
#define NEGV (-1e30f)

// ---- CDNA5 async global->LDS gather (gfx1250 only) ------------------------
// GV mode: 64-bit global address in a VGPR pair, LDS byte offset in a VGPR.
// Tracked by ASYNCcnt; completes in order vs other async loads of this wave.
__device__ __forceinline__ void async_gather_b32(const float* gaddr, uint32_t lds_off) {
    asm volatile("global_load_async_to_lds_b32 %0, %1, off"
                 :
                 : "v"(lds_off), "v"(gaddr)
                 : "memory");
}

#if __has_builtin(__builtin_amdgcn_s_wait_asynccnt)
#define WAIT_ASYNCCNT(n) __builtin_amdgcn_s_wait_asynccnt(n)
#else
#define WAIT_ASYNCCNT(n) asm volatile("s_wait_asynccnt %0" ::"n"(n) : "memory")
#endif

// Generic pointer to LDS: low 32 bits are the workgroup-relative LDS byte
// address (SHARED aperture layout), which is what the async VDST VGPR wants.
__device__ __forceinline__ uint32_t lds_off_of(const void* p) {
    return (uint32_t)(uintptr_t)p;
}

// log(exp(a-m)+exp(b-m)+exp(c-m)) + m, m = max(a,b,c); matches reference exactly.
__device__ __forceinline__ float lse3(float a, float b, float c) {
    const float m = fmaxf(a, fmaxf(b, c));
    return m + logf(expf(a - m) + expf(b - m) + expf(c - m));
}

__global__ __launch_bounds__(NTH) void ctc_alpha_kernel(
    const float* __restrict__ lp,      // [B, T, V] log-probs
    const int*   __restrict__ toks,    // [B, S]
    const int*   __restrict__ tlens,   // [B] cqt_lens
    const int*   __restrict__ slens,   // [B] token_lens_gt
    float*       __restrict__ loss_out // [B] loss_b / Sl_b
) {
    __shared__ float s_emit[NBUF][NTH];   // async-gathered emissions (states 0..511)
    __shared__ float s_alpha[2][L_ + 3];  // double-buffered alpha, states 0..512

    const int tid = threadIdx.x;
    const int b   = blockIdx.x;
    const int Tl  = tlens[b];             // >= T/2 >= 1024, so loop always runs

    // ext[s] = (s odd) ? tok[s/2] : BLANK(0). Thread `tid` owns state `tid`;
    // thread 0 additionally owns state 512 (blank, same emission column 0).
    int col = 0;
    int repok = 0;
    if (tid & 1) {
        const int k = tid >> 1;
        col = toks[b * S_ + k];                          // in [1, V)
        repok = (tid == 1) ? 1 : (col != toks[b * S_ + k - 1]);
    }

    const float* gbase = lp + (size_t)b * T_ * V_ + col; // column base; +t*V per step

    // ---- Prologue: prefetch emissions for t = 0 .. NBUF-1 ----
    for (int t = 0; t < NBUF; ++t) {
        async_gather_b32(gbase + (size_t)t * V_, lds_off_of(&s_emit[t][tid]));
    }

    // ---- t = 0: init alpha ----
    WAIT_ASYNCCNT(NBUF - 1);              // oldest (t=0) gather done
    const float e0 = s_emit[0][tid];
    float a = (tid <= 1) ? e0 : NEGV;     // alpha[0]=lp[0,blank], alpha[1]=lp[0,tok0]
    float a512 = NEGV;                    // thread 0's second state
    int p = 0;
    s_alpha[0][tid] = a;
    if (tid == 0) s_alpha[0][L_ - 1] = NEGV;
    {   // refill slot 0 with t = NBUF (clamped if beyond Tl-1)
        const int tp = NBUF;
        const int tc = (tp < Tl) ? tp : (Tl - 1);
        async_gather_b32(gbase + (size_t)tc * V_, lds_off_of(&s_emit[tp & (NBUF - 1)][tid]));
    }
    WAIT_ASYNCCNT(NBUF - 1);              // t=1 gather done (in-order)
    float e_next = s_emit[1 & (NBUF - 1)][tid];
    __syncthreads();

    // ---- Main DP over time: one barrier per step ----
    for (int t = 1; t < Tl; ++t) {
        const float e  = e_next;          // preloaded last iteration (pre-barrier)
        const float a1 = (tid >= 1) ? s_alpha[p][tid - 1] : NEGV;
        const float a2 = (repok) ? s_alpha[p][tid - 2] : NEGV;  // repok==0 for tid 0, 1 for tid 1

        const float nv = e + lse3(a, a1, a2);
        a = nv;
        s_alpha[p ^ 1][tid] = nv;

        if (tid == 0) {                   // state 512: blank, emission == e (col 0)
            const float b1 = s_alpha[p][L_ - 2];
            const float nv512 = e + lse3(a512, b1, NEGV);
            a512 = nv512;
            s_alpha[p ^ 1][L_ - 1] = nv512;
        }

        // Prefetch t+NBUF into the slot just consumed (clamped at the tail so
        // exactly NBUF async ops stay in flight and the wait immediate is valid).
        const int tp = t + NBUF;
        const int tc = (tp < Tl) ? tp : (Tl - 1);
        async_gather_b32(gbase + (size_t)tc * V_, lds_off_of(&s_emit[tp & (NBUF - 1)][tid]));

        // Preload e(t+1) so the wait + DS read overlap the barrier.
        WAIT_ASYNCCNT(NBUF - 1);          // oldest outstanding (t+1) complete
        e_next = s_emit[(t + 1) & (NBUF - 1)][tid];

        __syncthreads();
        p ^= 1;
    }

    // ---- Finalize: logsumexp(alpha[2*Sl], alpha[2*Sl-1]) ----
    if (tid == 0) {
        const int Sl = slens[b];
        const float lA = s_alpha[p][2 * Sl];
        const float lB = s_alpha[p][2 * Sl - 1];
        const float m  = fmaxf(lA, lB);
        const float ll = m + logf(expf(lA - m) + expf(lB - m));
        float loss = -ll;
        if (loss > 1e29f) loss = 0.0f;
        loss_out[b] = loss / (float)Sl;
    }
}

// Deterministic fixed-order reduction (no float atomics).
__global__ void ctc_reduce_kernel(const float* __restrict__ per_b, float* __restrict__ out) {
    if (threadIdx.x == 0 && blockIdx.x == 0) {
        float s = 0.0f;
        for (int i = 0; i < B_; ++i) s += per_b[i];
        out[0] = s / (float)B_;
    }
}

extern "C" void kernel_launch(void* const* d_in, const int* in_sizes, int n_in,
                              void* d_out, int out_size, void* d_ws, size_t ws_size,
                              hipStream_t stream) {
    (void)in_sizes; (void)n_in; (void)out_size; (void)ws_size;
    const float* lp    = (const float*)d_in[0];   // preds_logprob [B,T,V] f32
    const int*   toks  = (const int*)  d_in[1];   // tokens_gt     [B,S]  i32
    const int*   tlens = (const int*)  d_in[2];   // cqt_lens      [B]    i32
    const int*   slens = (const int*)  d_in[3];   // token_lens_gt [B]    i32
    float* ws  = (float*)d_ws;                    // [B] per-sample normalized losses
    float* out = (float*)d_out;                   // scalar f32

    ctc_alpha_kernel<<<dim3(B_), dim3(NTH), 0, stream>>>(lp, toks, tlens, slens, ws);
    ctc_reduce_kernel<<<dim3(1), dim3(32), 0, stream>>>(ws, out);
}